// gcn_air_60559038874089
// MI455X (gfx1250) — compile-verified
//
#include <hip/hip_runtime.h>
#include <math.h>

#define NNODES 50000
#define EEDGES 800000
#define NHID   128
#define NCLASS 40
#define HOPS   4
#define NCPAD  48   // NCLASS padded to 3 WMMA n-tiles

typedef float v2f __attribute__((ext_vector_type(2)));
typedef float v8f __attribute__((ext_vector_type(8)));

// ---------------------------------------------------------------- zero
__global__ void k_zero(float* __restrict__ p, long n) {
  long i = (long)blockIdx.x * blockDim.x + threadIdx.x;
  if (i < n) p[i] = 0.f;
}

// ---------------------------------------------------------------- SpMM (edge-parallel, L2 atomics)
// 2 edges per 256-thread block; 128 lanes per edge -> coalesced gather + atomic scatter.
__global__ void k_spmm(const int* __restrict__ row, const int* __restrict__ col,
                       const float* __restrict__ val, const float* __restrict__ x,
                       float* __restrict__ y) {
  int e = blockIdx.x * 2 + (threadIdx.x >> 7);
  int f = threadIdx.x & 127;
  if (e < EEDGES) {
    int r = row[e], c = col[e];
    float v = val[e];
    atomicAdd(&y[(long)r * NHID + f], v * x[(long)c * NHID + f]);
  }
}

// ---------------------------------------------------------------- WMMA fp32 GEMM: Y[M,NOUT] = X[M,128] * W[128,NOUT]
// One 16-row tile per block; wave w owns n-tile w. B staged transposed in LDS.
template <int NOUT, bool RELU>
__global__ void k_gemm(const float* __restrict__ X, const float* __restrict__ W,
                       float* __restrict__ Y) {
  constexpr int AP = 132;                       // even pitch: 8B-aligned b64 loads, 4-bank row skew
  __shared__ float As[16 * AP];                 // A tile, row-major
  __shared__ float Bt[NOUT * AP];               // Bt[n][k] = W[k*NOUT + n]
  const int tid = threadIdx.x;
  const int m0  = blockIdx.x * 16;

  for (int i = tid; i < 16 * 128; i += blockDim.x) {
    int r = i >> 7, c = i & 127;
    As[r * AP + c] = X[(long)(m0 + r) * 128 + c];
  }
  for (int i = tid; i < 128 * NOUT; i += blockDim.x) {
    int k = i / NOUT, n = i % NOUT;
    Bt[n * AP + k] = W[i];
  }
  __syncthreads();

  const int lane = tid & 31;
  const int wv   = tid >> 5;
  if (wv < NOUT / 16) {                          // wave-uniform: EXEC stays all-ones for WMMA
    const int mrow  = lane & 15;
    const int khalf = (lane >> 4) << 1;          // 0 or 2
    const int ncol  = wv * 16 + (lane & 15);
    v8f acc = {0.f, 0.f, 0.f, 0.f, 0.f, 0.f, 0.f, 0.f};
#pragma unroll
    for (int k = 0; k < 128; k += 4) {
      v2f a = *(const v2f*)&As[mrow * AP + k + khalf];   // A[m][k+khalf .. +1]
      v2f b = *(const v2f*)&Bt[ncol * AP + k + khalf];   // B[k+khalf][n], B[k+khalf+1][n]
      acc = __builtin_amdgcn_wmma_f32_16x16x4_f32(false, a, false, b,
                                                  (short)0, acc, false, false);
    }
#pragma unroll
    for (int j = 0; j < 8; ++j) {
      float v = acc[j];
      if (RELU) v = fmaxf(v, 0.f);
      int rrow = m0 + j + ((lane >> 4) << 3);            // M = j + 8*(lane/16)
      Y[(long)rrow * NOUT + ncol] = v;
    }
  }
}

// ---------------------------------------------------------------- pad fc_w (128x40 -> 128x48, zero-filled)
__global__ void k_pad_fcw(const float* __restrict__ fcw, float* __restrict__ out) {
  int i = blockIdx.x * blockDim.x + threadIdx.x;
  if (i < 128 * NCPAD) {
    int k = i / NCPAD, n = i % NCPAD;
    out[i] = (n < NCLASS) ? fcw[k * NCLASS + n] : 0.f;
  }
}

// ---------------------------------------------------------------- hop attention: wave per node
// scores use all 4 hop embeddings; output combination matches the reference exactly:
//   output = (W0+W1)*h_list[1] + W2*h_list[2] + W3*h_list[3]   (h_list[4] unused in output)
__global__ void k_att(const float* __restrict__ feat,
                      const float* __restrict__ h0, const float* __restrict__ h1,
                      const float* __restrict__ h2, const float* __restrict__ h3,
                      const float* __restrict__ attw, const float* __restrict__ attb,
                      float* __restrict__ out) {
  int node = blockIdx.x * 8 + (threadIdx.x >> 5);
  int lane = threadIdx.x & 31;
  long base = (long)node * 128 + (lane << 2);
  float4 xf = *(const float4*)&feat[base];
  float4 a0 = *(const float4*)&h0[base];
  float4 a1 = *(const float4*)&h1[base];
  float4 a2 = *(const float4*)&h2[base];
  float4 a3 = *(const float4*)&h3[base];
  float4 wh = *(const float4*)&attw[lane << 2];
  float4 wf = *(const float4*)&attw[128 + (lane << 2)];

  float pf = xf.x*wf.x + xf.y*wf.y + xf.z*wf.z + xf.w*wf.w;
  float p0 = a0.x*wh.x + a0.y*wh.y + a0.z*wh.z + a0.w*wh.w;
  float p1 = a1.x*wh.x + a1.y*wh.y + a1.z*wh.z + a1.w*wh.w;
  float p2 = a2.x*wh.x + a2.y*wh.y + a2.z*wh.z + a2.w*wh.w;
  float p3 = a3.x*wh.x + a3.y*wh.y + a3.z*wh.z + a3.w*wh.w;
  for (int off = 16; off; off >>= 1) {
    pf += __shfl_xor(pf, off, 32);
    p0 += __shfl_xor(p0, off, 32);
    p1 += __shfl_xor(p1, off, 32);
    p2 += __shfl_xor(p2, off, 32);
    p3 += __shfl_xor(p3, off, 32);
  }
  float b  = attb[0];
  float s0 = 1.f / (1.f + __expf(-(p0 + pf + b)));
  float s1 = 1.f / (1.f + __expf(-(p1 + pf + b)));
  float s2 = 1.f / (1.f + __expf(-(p2 + pf + b)));
  float s3 = 1.f / (1.f + __expf(-(p3 + pf + b)));
  float m  = fmaxf(fmaxf(s0, s1), fmaxf(s2, s3));
  float e0 = __expf(s0 - m), e1 = __expf(s1 - m), e2 = __expf(s2 - m), e3 = __expf(s3 - m);
  float inv = 1.f / (e0 + e1 + e2 + e3);
  float w01 = (e0 + e1) * inv;   // W[0] + W[1], both multiply h_list[1]
  float w2  = e2 * inv;
  float w3  = e3 * inv;

  float4 o;
  o.x = w01*a0.x + w2*a1.x + w3*a2.x;
  o.y = w01*a0.y + w2*a1.y + w3*a2.y;
  o.z = w01*a0.z + w2*a1.z + w3*a2.z;
  o.w = w01*a0.w + w2*a1.w + w3*a2.w;
  *(float4*)&out[base] = o;
}

// ---------------------------------------------------------------- log_softmax over 40 classes: wave per node
__global__ void k_logsm(const float* __restrict__ logits, const float* __restrict__ fcb,
                        float* __restrict__ out) {
  int node = blockIdx.x * 8 + (threadIdx.x >> 5);
  int lane = threadIdx.x & 31;
  const float NEG = -3.0e38f;
  long base = (long)node * NCPAD;
  float l0 = (lane < NCLASS)      ? logits[base + lane]      + fcb[lane]      : NEG;
  float l1 = (lane + 32 < NCLASS) ? logits[base + lane + 32] + fcb[lane + 32] : NEG;
  float m = fmaxf(l0, l1);
  for (int off = 16; off; off >>= 1) m = fmaxf(m, __shfl_xor(m, off, 32));
  float s = ((lane < NCLASS)      ? __expf(l0 - m) : 0.f) +
            ((lane + 32 < NCLASS) ? __expf(l1 - m) : 0.f);
  for (int off = 16; off; off >>= 1) s += __shfl_xor(s, off, 32);
  float lse = m + __logf(s);
  if (lane < NCLASS)      out[(long)node * NCLASS + lane]      = l0 - lse;
  if (lane + 32 < NCLASS) out[(long)node * NCLASS + lane + 32] = l1 - lse;
}

// ---------------------------------------------------------------- launcher
extern "C" void kernel_launch(void* const* d_in, const int* in_sizes, int n_in,
                              void* d_out, int out_size, void* d_ws, size_t ws_size,
                              hipStream_t stream) {
  const float* feat = (const float*)d_in[0];
  const int*   erow = (const int*)d_in[1];
  const int*   ecol = (const int*)d_in[2];
  const float* eval = (const float*)d_in[3];
  const float* Wg   = (const float*)d_in[4];
  const float* attw = (const float*)d_in[5];
  const float* attb = (const float*)d_in[6];
  const float* fcw  = (const float*)d_in[7];
  const float* fcb  = (const float*)d_in[8];
  float* out = (float*)d_out;

  float* ws = (float*)d_ws;
  const long NF = (long)NNODES * NHID;          // 6.4M floats per buffer
  float* h[HOPS];
  for (int i = 0; i < HOPS; ++i) h[i] = ws + (long)i * NF;
  float* spmm   = ws + 4 * NF;                  // scratch; reused as att_out after hops
  float* attout = spmm;
  float* fcwp   = ws + 5 * NF;                  // 128x48 padded fc_w
  float* logits = fcwp + 128 * NCPAD;           // N x 48

  const float* xin = feat;
  for (int i = 0; i < HOPS; ++i) {
    k_zero<<<(int)((NF + 255) / 256), 256, 0, stream>>>(spmm, NF);
    k_spmm<<<EEDGES / 2, 256, 0, stream>>>(erow, ecol, eval, xin, spmm);
    k_gemm<NHID, true><<<NNODES / 16, 256, 0, stream>>>(spmm, Wg + (long)i * NHID * NHID, h[i]);
    xin = h[i];
  }
  k_pad_fcw<<<(128 * NCPAD + 255) / 256, 256, 0, stream>>>(fcw, fcwp);
  k_att<<<NNODES / 8, 256, 0, stream>>>(feat, h[0], h[1], h[2], h[3], attw, attb, attout);
  k_gemm<NCPAD, false><<<NNODES / 16, 128, 0, stream>>>(attout, fcwp, logits);
  k_logsm<<<NNODES / 8, 256, 0, stream>>>(logits, fcb, out);
}